// GNNEncoder_38474317038224
// MI455X (gfx1250) — compile-verified
//
#include <hip/hip_runtime.h>
#include <hip/hip_bf16.h>

typedef float v2f __attribute__((ext_vector_type(2)));
typedef float v8f __attribute__((ext_vector_type(8)));

#define DIM 256
#define HEADS 4
#define HC 64
#define TSTEP 20
#define NEG_SLOPE_C 0.2f
#define PROX2_C 2500.0f   // 50^2 : compare squared distance, no sqrt needed

// ---------------------------------------------------------------------------
// Kernel 1: node feature extraction + linear + relu + LayerNorm, and pos[]
// One block (256 threads) per node; thread d owns channel d.
// ---------------------------------------------------------------------------
__global__ void embed_kernel(const float* __restrict__ ego,
                             const float* __restrict__ nbr,
                             const float* __restrict__ node_w, const float* __restrict__ node_b,
                             const float* __restrict__ node_g, const float* __restrict__ node_be,
                             const float* __restrict__ ego_w,  const float* __restrict__ ego_b,
                             const float* __restrict__ ego_g,  const float* __restrict__ ego_be,
                             float* __restrict__ h, float* __restrict__ pos,
                             int N) {
  int node = blockIdx.x;
  int b = node / N, n = node % N;
  int d = threadIdx.x;
  __shared__ float xs[5];
  __shared__ float s1[DIM], s2[DIM];
  if (d < 5) {
    float v;
    if (n == 0) v = ego[(b * TSTEP + (TSTEP - 1)) * 7 + d];
    else        v = nbr[(((long)b * (N - 1) + (n - 1)) * TSTEP + (TSTEP - 1)) * 11 + d];
    xs[d] = v;
  }
  __syncthreads();
  if (d == 0) { pos[node * 2 + 0] = xs[0]; pos[node * 2 + 1] = xs[1]; }
  const float *W, *bb, *g, *be;
  if (n == 0) { W = ego_w;  bb = ego_b;  g = ego_g;  be = ego_be; }
  else        { W = node_w; bb = node_b; g = node_g; be = node_be; }
  // only the first 5 padded features are nonzero
  float y = bb[d];
#pragma unroll
  for (int k = 0; k < 5; ++k) y = fmaf(xs[k], W[k * DIM + d], y);
  y = fmaxf(y, 0.0f);
  s1[d] = y; s2[d] = y * y;
  __syncthreads();
  for (int off = DIM / 2; off > 0; off >>= 1) {
    if (d < off) { s1[d] += s1[d + off]; s2[d] += s2[d + off]; }
    __syncthreads();
  }
  float mean = s1[0] * (1.0f / DIM);
  float var  = s2[0] * (1.0f / DIM) - mean * mean;
  h[(long)node * DIM + d] = (y - mean) * rsqrtf(var + 1e-5f) * g[d] + be[d];
}

// ---------------------------------------------------------------------------
// Kernel 2: fp32 WMMA GEMM  C[M,N] = A[M,K] @ B[K,N] (+ bias)
// block = 256 threads = 8 waves; wave w -> 16-row tile, 64 output cols/block.
// Uses V_WMMA_F32_16X16X4_F32 (exact fp32).
// ---------------------------------------------------------------------------
__global__ void gemm_wmma_f32(const float* __restrict__ A,
                              const float* __restrict__ Bm,
                              const float* __restrict__ bias,  // may be null
                              float* __restrict__ C,
                              int M, int K, int N) {
  int wave = threadIdx.x >> 5;
  int lane = threadIdx.x & 31;
  int m0 = (blockIdx.x * 8 + wave) * 16;
  int n0 = blockIdx.y * 64;
  if (m0 >= M) return;                 // wave-uniform: EXEC stays all-ones
  int lhi = lane >> 4;                 // 0: lanes 0-15, 1: lanes 16-31
  int l15 = lane & 15;
  v8f acc[4];
  acc[0] = 0.f; acc[1] = 0.f; acc[2] = 0.f; acc[3] = 0.f;
  const float* arow = A + (long)(m0 + l15) * K;
  for (int k0 = 0; k0 < K; k0 += 4) {
    // A 16x4: lanes 0-15 hold K=k0,k0+1 ; lanes 16-31 hold K=k0+2,k0+3
    v2f af;
    af.x = arow[k0 + lhi * 2 + 0];
    af.y = arow[k0 + lhi * 2 + 1];
#pragma unroll
    for (int t = 0; t < 4; ++t) {
      int nb = n0 + t * 16 + l15;
      // B 4x16: v0 -> rows K=k0 (lanes 0-15) / k0+2 (lanes 16-31); v1 -> k0+1 / k0+3
      v2f bf;
      bf.x = Bm[(long)(k0 + lhi * 2 + 0) * N + nb];
      bf.y = Bm[(long)(k0 + lhi * 2 + 1) * N + nb];
      acc[t] = __builtin_amdgcn_wmma_f32_16x16x4_f32(
          false, af, false, bf, (short)0, acc[t], false, false);
    }
  }
#pragma unroll
  for (int t = 0; t < 4; ++t) {
    int col = n0 + t * 16 + l15;
    float bv = bias ? bias[col] : 0.0f;
#pragma unroll
    for (int g = 0; g < 8; ++g) {
      int row = m0 + g + lhi * 8;      // C/D: VGPR g holds rows g and g+8
      C[(long)row * N + col] = acc[t][g] + bv;
    }
  }
}

// ---------------------------------------------------------------------------
// Kernel 3: per-node attention coefficients a_s, a_d (dot over 64 channels)
// ---------------------------------------------------------------------------
__global__ void attn_coef(const float* __restrict__ wh,
                          const float* __restrict__ asrc,
                          const float* __restrict__ adst,
                          float* __restrict__ a_s, float* __restrict__ a_d,
                          int total) {
  int idx = blockIdx.x * blockDim.x + threadIdx.x;
  if (idx >= total) return;
  int hh = idx & (HEADS - 1);
  long node = idx >> 2;
  const float* w = wh + node * DIM + hh * HC;
  float ss = 0.f, dd = 0.f;
#pragma unroll 8
  for (int c = 0; c < HC; ++c) {
    float v = w[c];
    ss = fmaf(v, asrc[hh * HC + c], ss);
    dd = fmaf(v, adst[hh * HC + c], dd);
  }
  a_s[idx] = ss;
  a_d[idx] = dd;
}

// ---------------------------------------------------------------------------
// Kernel 4: fused GAT attention aggregation, WMMA phase-2.
// Block = (b, 16-row tile), 256 threads = 8 waves.
// Phase 1 (per 16-j chunk): 256 threads compute p[h][16i][16j] = masked
// exp(leakyrelu(a_s[j]+a_d[i])) ONCE into LDS (shared by the 64 channels of
// each head); logits are O(1) so direct exp == softmax after the final divide.
// Phase 2: each wave owns two 16x16 output column-tiles (both in head w>>1)
// and does P(16x16, LDS) @ wh(16x16) with V_WMMA_F32_16X16X4_F32.
// Epilogue: divide by softmax denominator, +bias, relu, store.
// ---------------------------------------------------------------------------
__global__ void gat_aggregate(const float* __restrict__ wh,
                              const float* __restrict__ a_s,
                              const float* __restrict__ a_d,
                              const float* __restrict__ pos,
                              const float* __restrict__ bias,   // gat_b[l]
                              float* __restrict__ h_out,
                              int N) {
  int ntiles = N >> 4;
  int b  = blockIdx.x / ntiles;
  int i0 = (blockIdx.x % ntiles) << 4;
  int t = threadIdx.x;
  int wave = t >> 5;
  int lane = t & 31;
  int lhi  = lane >> 4;
  int l15  = lane & 15;

  __shared__ float posi[16][2];
  __shared__ float adi[16][HEADS];
  __shared__ float posj[16][2];
  __shared__ float asj[16][HEADS];
  __shared__ float plds[HEADS][16][18];   // padded: kills i*16 bank conflicts
  __shared__ float sred[4][16][HEADS];
  __shared__ float sden[16][HEADS];

  if (t < 32)            { int i = t >> 1, c = t & 1; posi[i][c] = pos[((long)b * N + i0 + i) * 2 + c]; }
  else if (t < 96)       { int q = t - 32; int i = q >> 2, hh = q & 3; adi[i][hh] = a_d[((long)b * N + i0 + i) * HEADS + hh]; }

  // phase-1 mapping: 256 threads = 16 i x 4 heads x 4 j-groups (4 j each)
  int i1 = t & 15, h1 = (t >> 4) & 3, jg = t >> 6;
  // phase-2 mapping: wave -> column tiles (2w, 2w+1); head hW uniform per wave
  int hW  = wave >> 1;
  int c0  = (wave * 2) * 16 + l15;   // col of tile 0
  int c1  = c0 + 16;                 // col of tile 1

  v8f acc0, acc1;
  acc0 = 0.f; acc1 = 0.f;
  float sp = 0.f;

  for (int j0 = 0; j0 < N; j0 += 16) {
    __syncthreads();   // previous phase-2 reads of plds / first-iter posi,adi
    if (t < 32)                 { int jj = t >> 1, cc = t & 1; posj[jj][cc] = pos[((long)b * N + j0 + jj) * 2 + cc]; }
    else if (t >= 64 && t < 128){ int q = t - 64; int jj = q >> 2, hh = q & 3; asj[jj][hh] = a_s[((long)b * N + j0 + jj) * HEADS + hh]; }
    __syncthreads();
    // ---- phase 1: p weights (computed once, shared per head) ----
#pragma unroll
    for (int u = 0; u < 4; ++u) {
      int jj = jg * 4 + u;
      float dx = posi[i1][0] - posj[jj][0];
      float dy = posi[i1][1] - posj[jj][1];
      bool adj = (dx * dx + dy * dy < PROX2_C) || ((i0 + i1) == (j0 + jj));
      float e = asj[jj][h1] + adi[i1][h1];
      e = e > 0.f ? e : NEG_SLOPE_C * e;         // leaky_relu(0.2)
      float p = adj ? __expf(e) : 0.f;
      plds[h1][i1][jj] = p;
      sp += p;
    }
    __syncthreads();
    // ---- phase 2: WMMA  out_tile += P(16x16) @ wh_chunk(16x16/tile) ----
    const float* whj = wh + ((long)b * N + j0) * DIM;
#pragma unroll
    for (int ks = 0; ks < 16; ks += 4) {
      int kk = ks + lhi * 2;
      // A fragment from LDS: lane row = l15, K pair = (kk, kk+1)
      v2f af;
      af.x = plds[hW][l15][kk + 0];
      af.y = plds[hW][l15][kk + 1];
      // B fragments from wh: rows j0+kk, j0+kk+1 ; cols c0 / c1
      v2f bf0, bf1;
      bf0.x = whj[(long)(kk + 0) * DIM + c0];
      bf0.y = whj[(long)(kk + 1) * DIM + c0];
      bf1.x = whj[(long)(kk + 0) * DIM + c1];
      bf1.y = whj[(long)(kk + 1) * DIM + c1];
      acc0 = __builtin_amdgcn_wmma_f32_16x16x4_f32(
          false, af, false, bf0, (short)0, acc0, false, false);
      acc1 = __builtin_amdgcn_wmma_f32_16x16x4_f32(
          false, af, false, bf1, (short)0, acc1, false, false);
    }
  }
  // reduce softmax denominators over the 4 j-groups
  sred[jg][i1][h1] = sp;
  __syncthreads();
  if (t < 64) {
    int i = t >> 2, hh = t & 3;
    sden[i][hh] = sred[0][i][hh] + sred[1][i][hh] + sred[2][i][hh] + sred[3][i][hh];
  }
  __syncthreads();
  // epilogue on WMMA C fragments: row = g + lhi*8, col = c0/c1
  float bv0 = bias[c0];
  float bv1 = bias[c1];
#pragma unroll
  for (int g = 0; g < 8; ++g) {
    int row = g + lhi * 8;
    float inv = 1.0f / sden[row][hW];
    long base = ((long)b * N + i0 + row) * DIM;
    h_out[base + c0] = fmaxf(acc0[g] * inv + bv0, 0.f);
    h_out[base + c1] = fmaxf(acc1[g] * inv + bv1, 0.f);
  }
}

// ---------------------------------------------------------------------------
// Kernel 5: zero the mask tail of d_out
// ---------------------------------------------------------------------------
__global__ void zero_tail(float* __restrict__ out, long start, long end) {
  long i = (long)blockIdx.x * blockDim.x + threadIdx.x + start;
  if (i < end) out[i] = 0.f;
}

extern "C" void kernel_launch(void* const* d_in, const int* in_sizes, int n_in,
                              void* d_out, int out_size, void* d_ws, size_t ws_size,
                              hipStream_t stream) {
  const float* ego      = (const float*)d_in[0];
  const float* nbr      = (const float*)d_in[1];
  const float* node_w   = (const float*)d_in[2];
  const float* node_b   = (const float*)d_in[3];
  const float* node_g   = (const float*)d_in[4];
  const float* node_be  = (const float*)d_in[5];
  const float* ego_w    = (const float*)d_in[6];
  const float* ego_b    = (const float*)d_in[7];
  const float* ego_g    = (const float*)d_in[8];
  const float* ego_be   = (const float*)d_in[9];
  const float* gat_w    = (const float*)d_in[10];
  const float* gat_asrc = (const float*)d_in[11];
  const float* gat_adst = (const float*)d_in[12];
  const float* gat_b    = (const float*)d_in[13];
  const float* proj_w   = (const float*)d_in[14];
  const float* proj_b   = (const float*)d_in[15];

  int B  = in_sizes[0] / (TSTEP * 7);
  int Nn = in_sizes[1] / (B * TSTEP * 11);
  int N  = Nn + 1;                     // 1024
  long BN = (long)B * N;               // 16384

  float* ws  = (float*)d_ws;
  float* pos = ws;                     // BN*2
  float* a_s = pos + BN * 2;           // BN*4
  float* a_d = a_s + BN * HEADS;       // BN*4
  float* h   = a_d + BN * HEADS;       // BN*256
  float* wh  = h + BN * DIM;           // BN*256

  embed_kernel<<<(int)BN, DIM, 0, stream>>>(ego, nbr, node_w, node_b, node_g, node_be,
                                            ego_w, ego_b, ego_g, ego_be, h, pos, N);

  dim3 ggrid((unsigned)(BN / 128), DIM / 64);
  int total = (int)(BN * HEADS);
  for (int l = 0; l < 2; ++l) {
    const float* Wl = gat_w + (long)l * DIM * DIM;
    gemm_wmma_f32<<<ggrid, 256, 0, stream>>>(h, Wl, nullptr, wh, (int)BN, DIM, DIM);
    attn_coef<<<(total + 255) / 256, 256, 0, stream>>>(wh,
        gat_asrc + (long)l * HEADS * HC, gat_adst + (long)l * HEADS * HC, a_s, a_d, total);
    gat_aggregate<<<B * (N / 16), 256, 0, stream>>>(wh, a_s, a_d, pos,
        gat_b + (long)l * DIM, h, N);
  }

  // final projection straight into d_out
  gemm_wmma_f32<<<ggrid, 256, 0, stream>>>(h, proj_w, proj_b, (float*)d_out,
                                           (int)BN, DIM, DIM);

  long start = BN * DIM;
  long rem = (long)out_size - start;
  if (rem > 0)
    zero_tail<<<(int)((rem + 255) / 256), 256, 0, stream>>>((float*)d_out, start, (long)out_size);
}